// MNEMatch_44822278701595
// MI455X (gfx1250) — compile-verified
//
#include <hip/hip_runtime.h>

typedef float v2f __attribute__((ext_vector_type(2)));
typedef float v8f __attribute__((ext_vector_type(8)));

#define N1 128
#define N2 128
#define DDIM 512
#define SSTRIDE 132      // S row stride (floats)
#define KC 32            // K-chunk (floats) = 128 B per row
#define NCHUNK (DDIM / KC)
#define BSTRIDE 36       // staged-chunk row stride (floats): 144 B, 16B-aligned, conflict-free

__global__ __launch_bounds__(256) void mne_match_kernel(const float* __restrict__ x1,
                                                        const float* __restrict__ x2,
                                                        float* __restrict__ out) {
    __shared__ float S[N1 * SSTRIDE];            // 67,584 B
    __shared__ float Bbuf[2][N2 * BSTRIDE];      // 2 x 18,432 B staging for x2 chunks
    __shared__ float rowMaxL[N1];
    __shared__ int   rowArgL[N1];

    const int b    = blockIdx.x;
    const int tid  = threadIdx.x;
    const int wave = tid >> 5;
    const int lane = tid & 31;
    const int m    = lane & 15;        // row/col within 16-wide fragment
    const int h    = lane >> 4;        // lane half selects K sub-pair

    const float* A  = x1 + (size_t)b * N1 * DDIM;   // [128][512]
    const float* Bm = x2 + (size_t)b * N2 * DDIM;   // [128][512]

    // byte offset of Bbuf within LDS (generic LDS addr low 32 bits == LDS offset)
    const unsigned ldsBase = (unsigned)(uintptr_t)(&Bbuf[0][0]);

    // Issue one chunk of x2 into LDS buffer tb via async copy:
    // 1024 x 16B segments, 4 per thread, coalesced 128B per row.
    auto issue_chunk = [&](int cc, int tb) {
#pragma unroll
        for (int i = 0; i < 4; ++i) {
            int seg = tid + 256 * i;                 // 0..1023
            int row = seg >> 3;                      // 0..127
            int sc  = seg & 7;                       // 16B segment within 128B row slice
            const float* g = Bm + (size_t)row * DDIM + KC * cc + sc * 4;
            unsigned l = ldsBase + (unsigned)(tb * (N2 * BSTRIDE * 4) + (row * BSTRIDE + sc * 4) * 4);
            asm volatile("global_load_async_to_lds_b128 %0, %1, off"
                         :: "v"(l), "v"(g) : "memory");
        }
    };

    // ---------------- Phase 1: S = A @ Bm^T via V_WMMA_F32_16X16X4_F32 ----------------
    // Wave w computes rows 16w..16w+15 of S for all 8 column tiles.
    // A fragments: direct global b64 (per-wave exclusive data, overlaps with WMMA).
    // B fragments: from LDS, staged by double-buffered async copies.
    const float* arow = A + (size_t)(16 * wave + m) * DDIM;
    v8f acc[8] = {};

    issue_chunk(0, 0);
    for (int c = 0; c < NCHUNK; ++c) {
        const int cur = c & 1;
        if (c + 1 < NCHUNK) {
            issue_chunk(c + 1, cur ^ 1);
            asm volatile("s_wait_asynccnt 0x4" ::: "memory");   // chunk c landed; c+1 in flight
        } else {
            asm volatile("s_wait_asynccnt 0x0" ::: "memory");
        }
        __syncthreads();                                        // chunk c visible to all waves

        const float* __restrict__ bch = &Bbuf[cur][0];
        const int k0 = KC * c;
        for (int kk = 0; kk < KC; kk += 4) {
            // A fragment 16x4: lanes 0-15 M=lane (K=k,k+1), lanes 16-31 M=lane-16 (K=k+2,k+3)
            v2f af = *(const v2f*)(arow + k0 + kk + 2 * h);
#pragma unroll
            for (int t = 0; t < 8; ++t) {
                // B fragment 4x16 of Bm^T from staged LDS: B[kk'][n] = x2[n][kk']
                v2f bf = *(const v2f*)(bch + (16 * t + m) * BSTRIDE + kk + 2 * h);
                acc[t] = __builtin_amdgcn_wmma_f32_16x16x4_f32(
                    false, af, false, bf, (short)0, acc[t], false, false);
            }
        }
        __syncthreads();    // everyone done reading buf[cur] before it is reloaded
    }

    // C/D layout: VGPR p -> M = p (lanes 0-15) or p+8 (lanes 16-31); N = lane&15
#pragma unroll
    for (int t = 0; t < 8; ++t) {
#pragma unroll
        for (int p = 0; p < 8; ++p) {
            int row = 16 * wave + p + 8 * h;
            int col = 16 * t + m;
            S[row * SSTRIDE + col] = acc[t][p];
        }
    }
    __syncthreads();

    // ---------------- Phase 2a: initial per-row (max, argmax), 128 threads ----------------
    if (tid < N1) {
        float bv = -3.0e38f; int bc = 0;
        const float* row = &S[tid * SSTRIDE];
        for (int c = 0; c < N2; ++c) {
            float v = row[c];
            if (v > bv) { bv = v; bc = c; }
        }
        rowMaxL[tid] = bv;
        rowArgL[tid] = bc;
    }
    __syncthreads();

    if (wave != 0) return;   // greedy phase is single-wave (no more block barriers)

    // ---------------- Phase 2b: greedy matching, wave 0 only ----------------
    // Lane owns rows lane*4 .. lane*4+3. Invariant: rmax[j]/rarg[j] is the max of
    // row lane*4+j over currently-active columns (recomputed when its column dies).
    float rmax[4]; int rarg[4];
#pragma unroll
    for (int j = 0; j < 4; ++j) {
        rmax[j] = rowMaxL[lane * 4 + j];
        rarg[j] = rowArgL[lane * 4 + j];
    }
    unsigned ract = 0xFu;                                // 4 active-row bits
    unsigned colMask[4] = {~0u, ~0u, ~0u, ~0u};          // replicated per lane
    float sum = 0.0f;

    for (int it = 0; it < 128; ++it) {
        // local best among this lane's active rows (tie-break: lowest flat index)
        float bv = -3.0e38f; int brow = 0, bcol = 0;
#pragma unroll
        for (int j = 0; j < 4; ++j) {
            if ((ract >> j) & 1u) {
                float v = rmax[j]; int rr = lane * 4 + j; int cc = rarg[j];
                if (v > bv || (v == bv && rr * 128 + cc < brow * 128 + bcol)) {
                    bv = v; brow = rr; bcol = cc;
                }
            }
        }
        // wave-wide butterfly reduce; every lane converges to the same winner
        for (int off = 16; off > 0; off >>= 1) {
            float ov  = __shfl_xor(bv, off, 32);
            int  orow = __shfl_xor(brow, off, 32);
            int  ocol = __shfl_xor(bcol, off, 32);
            if (ov > bv || (ov == bv && orow * 128 + ocol < brow * 128 + bcol)) {
                bv = ov; brow = orow; bcol = ocol;
            }
        }
        sum += bv;
        if ((brow >> 2) == lane) ract &= ~(1u << (brow & 3));
        colMask[bcol >> 5] &= ~(1u << (bcol & 31));

        // rescan only rows whose cached argmax column was just removed
#pragma unroll
        for (int j = 0; j < 4; ++j) {
            if (((ract >> j) & 1u) && rarg[j] == bcol) {
                float nv = -3.0e38f; int nc = 0;
                const float* row = &S[(lane * 4 + j) * SSTRIDE];
                for (int c = 0; c < N2; ++c) {
                    if ((colMask[c >> 5] >> (c & 31)) & 1u) {
                        float v = row[c];
                        if (v > nv) { nv = v; nc = c; }
                    }
                }
                rmax[j] = nv; rarg[j] = nc;
            }
        }
    }

    if (lane == 0) out[b] = sum * (1.0f / 128.0f);
}

extern "C" void kernel_launch(void* const* d_in, const int* in_sizes, int n_in,
                              void* d_out, int out_size, void* d_ws, size_t ws_size,
                              hipStream_t stream) {
    const float* x1 = (const float*)d_in[0];
    const float* x2 = (const float*)d_in[1];
    float* out = (float*)d_out;
    int B = in_sizes[0] / (N1 * DDIM);   // 512
    mne_match_kernel<<<B, 256, 0, stream>>>(x1, x2, out);
}